// MDGNN_57844619543216
// MI455X (gfx1250) — compile-verified
//
#include <hip/hip_runtime.h>
#include <hip/hip_bf16.h>

// ---------------------------------------------------------------------------
// MDGNN forward for MI455X (gfx1250), fp32 throughout using V_WMMA_F32_16X16X4_F32.
// Deterministic: no float atomics anywhere; CSR canonicalized by per-node sort;
// all reductions are fixed-order.
// Input flattening assumed to follow setup_inputs() dict insertion order
// recursively (x, edge_index, edge_attr, batch, then params leaves in order).
// ---------------------------------------------------------------------------

#define NN 50000
#define EE 400000
#define INDIM 64
#define EDDIM 16
#define DD 32
#define HH 3
#define TT 6
#define NBLK 3
#define BG 128
#define OUTD 12
#define MT 3125            // NN/16 row tiles
#define NB_GRU 391         // ceil(MT/8) blocks for fused GRU kernel

typedef float v2f __attribute__((ext_vector_type(2)));
typedef float v8f __attribute__((ext_vector_type(8)));

__device__ __forceinline__ v8f wmma4(v2f a, v2f b, v8f c) {
  // D = A(16x4 f32) * B(4x16 f32) + C(16x16 f32)
  return __builtin_amdgcn_wmma_f32_16x16x4_f32(false, a, false, b, (short)0, c,
                                               false, false);
}
__device__ __forceinline__ float sigf(float x) { return 1.0f / (1.0f + expf(-x)); }
__device__ __forceinline__ float celuf(float x) { return x > 0.0f ? x : expm1f(x); }

// ------------------------------ utility ------------------------------------
__global__ void k_zero_f(float* p, int n) {
  int i = blockIdx.x * blockDim.x + threadIdx.x;
  if (i < n) p[i] = 0.0f;
}
__global__ void k_zero_i(int* p, int n) {
  int i = blockIdx.x * blockDim.x + threadIdx.x;
  if (i < n) p[i] = 0;
}

// ------------------------------ CSR build ----------------------------------
__global__ void k_count(const int* __restrict__ ei, int* __restrict__ deg) {
  int e = blockIdx.x * blockDim.x + threadIdx.x;
  if (e < EE) atomicAdd(&deg[ei[EE + e]], 1);   // dst row
}

__global__ void k_scan(const int* __restrict__ deg, int* __restrict__ off,
                       int* __restrict__ cur) {
  __shared__ int s[1024];
  __shared__ int run;
  int tid = threadIdx.x;
  if (tid == 0) run = 0;
  __syncthreads();
  for (int base = 0; base < NN; base += 1024) {
    int idx = base + tid;
    int v = (idx < NN) ? deg[idx] : 0;
    s[tid] = v;
    __syncthreads();
    for (int o = 1; o < 1024; o <<= 1) {
      int t = (tid >= o) ? s[tid - o] : 0;
      __syncthreads();
      s[tid] += t;
      __syncthreads();
    }
    int excl = run + s[tid] - v;
    if (idx < NN) { off[idx] = excl; cur[idx] = excl; }
    __syncthreads();
    if (tid == 0) run += s[1023];
    __syncthreads();
  }
  if (tid == 0) off[NN] = run;
}

__global__ void k_fill(const int* __restrict__ ei, int* __restrict__ cur,
                       int* __restrict__ eid) {
  int e = blockIdx.x * blockDim.x + threadIdx.x;
  if (e < EE) {
    int pos = atomicAdd(&cur[ei[EE + e]], 1);
    eid[pos] = e;
  }
}

// canonical per-node order -> deterministic accumulation order every call
__global__ void k_sortcsr(const int* __restrict__ off, int* __restrict__ eid) {
  int v = blockIdx.x * blockDim.x + threadIdx.x;
  if (v >= NN) return;
  int s = off[v], e = off[v + 1];
  for (int i = s + 1; i < e; ++i) {
    int key = eid[i];
    int j = i - 1;
    while (j >= s && eid[j] > key) { eid[j + 1] = eid[j]; --j; }
    eid[j + 1] = key;
  }
}

__global__ void k_gstart(const int* __restrict__ batch, int* __restrict__ gs) {
  int b = threadIdx.x;
  if (b > BG) return;
  int lo = 0, hi = NN;
  while (lo < hi) {
    int mid = (lo + hi) >> 1;
    if (batch[mid] < b) lo = mid + 1; else hi = mid;
  }
  gs[b] = lo;
}

// ------------------------- generic WMMA fp32 GEMM ---------------------------
// C[16mt x 16nt] = A[.,K] @ B[K,.]   (row-major, K % 4 == 0, dims % 16 == 0)
// MODE 0: C = acc (+bias)     MODE 1: C = celu(acc + bias)     MODE 2: C += acc
template <int MODE>
__global__ void __launch_bounds__(256) k_gemm(const float* __restrict__ A, int lda,
                                              const float* __restrict__ B, int ldb,
                                              const float* __restrict__ bias,
                                              float* __restrict__ C, int ldc,
                                              int Mt, int Nt, int K) {
  int wave = threadIdx.x >> 5, lane = threadIdx.x & 31;
  int tile = blockIdx.x * 8 + wave;
  if (tile >= Mt * Nt) return;            // wave-uniform
  int mi = tile / Nt, ni = tile % Nt;
  int half = lane >> 4, l16 = lane & 15;
  v8f acc = {};
  if (MODE == 2) {
#pragma unroll
    for (int j = 0; j < 8; ++j)
      acc[j] = C[(size_t)(mi * 16 + j + 8 * half) * ldc + ni * 16 + l16];
  }
  const float* ap = A + (size_t)(mi * 16 + l16) * lda + 2 * half;
  const float* bp = B + (size_t)(2 * half) * ldb + ni * 16 + l16;
  for (int k = 0; k < K; k += 4) {
    v2f a; a.x = ap[k]; a.y = ap[k + 1];
    v2f b; b.x = bp[(size_t)k * ldb]; b.y = bp[(size_t)k * ldb + ldb];
    acc = wmma4(a, b, acc);
  }
  float bv = 0.0f;
  if (MODE != 2 && bias) bv = bias[ni * 16 + l16];
#pragma unroll
  for (int j = 0; j < 8; ++j) {
    float v = acc[j] + bv;
    if (MODE == 1) v = celuf(v);
    C[(size_t)(mi * 16 + j + 8 * half) * ldc + ni * 16 + l16] = v;
  }
}

// ----------------------------- small kernels --------------------------------
__global__ void k_binit(const float* __restrict__ x, float* __restrict__ xin,
                        float* __restrict__ h) {
  int i = blockIdx.x * blockDim.x + threadIdx.x;
  if (i < NN * DD) { float v = x[i]; xin[i] = v; h[i] = v; }
}

__global__ void k_wae(const float* __restrict__ we, const float* __restrict__ ae,
                      float* __restrict__ wae) {
  int t = threadIdx.x;
  if (t >= EDDIM * HH) return;
  int k = t / HH, h = t % HH;
  float s = 0.0f;
  for (int f = 0; f < DD; ++f) s = fmaf(we[k * 96 + h * 32 + f], ae[h * 32 + f], s);
  wae[t] = s;
}

__global__ void k_se(const float* __restrict__ ea, const float* __restrict__ wae,
                     float* __restrict__ se) {
  int id = blockIdx.x * blockDim.x + threadIdx.x;
  if (id >= EE * HH) return;
  int e = id / HH, h = id % HH;
  float s = 0.0f;
#pragma unroll
  for (int k = 0; k < EDDIM; ++k) s = fmaf(ea[e * EDDIM + k], wae[k * HH + h], s);
  se[id] = s;
}

__global__ void k_scores(const float* __restrict__ xp, const float* __restrict__ ai,
                         const float* __restrict__ aj, float* __restrict__ si,
                         float* __restrict__ sj) {
  int id = blockIdx.x * blockDim.x + threadIdx.x;
  if (id >= NN * HH) return;
  int n = id / HH, h = id % HH;
  const float* row = xp + (size_t)n * 96 + h * 32;
  const float* a = ai + h * 32;
  const float* b = aj + h * 32;
  float s1 = 0.0f, s2 = 0.0f;
#pragma unroll
  for (int f = 0; f < DD; ++f) { s1 = fmaf(row[f], a[f], s1); s2 = fmaf(row[f], b[f], s2); }
  si[id] = s1; sj[id] = s2;
}

// one wave per node: local softmax over in-edges for all 3 heads, no atomics.
// e_proj recomputed on the fly (16 FMAs from L0-resident we).
__global__ void __launch_bounds__(256) k_agg(
    const int* __restrict__ ei, const int* __restrict__ off,
    const int* __restrict__ eid, const float* __restrict__ si,
    const float* __restrict__ sj, const float* __restrict__ se,
    const float* __restrict__ xp, const float* __restrict__ ea,
    const float* __restrict__ we, const float* __restrict__ cb,
    float* __restrict__ m_out) {
  int wave = threadIdx.x >> 5, lane = threadIdx.x & 31;
  int v = blockIdx.x * 8 + wave;
  if (v >= NN) return;
  int s0 = off[v], d = off[v + 1] - s0;
  float msum = 0.0f;
  for (int h = 0; h < HH; ++h) {
    float siv = si[v * HH + h];
    float mx = -3.4e38f;
    for (int i = 0; i < d; ++i) {
      int e = eid[s0 + i];
      int u = ei[e];
      __builtin_prefetch(xp + (size_t)u * 96 + h * 32 + lane, 0, 1);
      float l = siv + sj[u * HH + h] + se[e * HH + h];
      l = (l >= 0.0f) ? l : 0.2f * l;
      mx = fmaxf(mx, l);
    }
    float ssum = 0.0f, acc = 0.0f;
    for (int i = 0; i < d; ++i) {
      int e = eid[s0 + i];
      int u = ei[e];
      float l = siv + sj[u * HH + h] + se[e * HH + h];
      l = (l >= 0.0f) ? l : 0.2f * l;
      float w = expf(l - mx);
      ssum += w;
      float ep = 0.0f;
      const float* war = we + h * 32 + lane;
      const float* ear = ea + (size_t)e * EDDIM;
#pragma unroll
      for (int k = 0; k < EDDIM; ++k) ep = fmaf(ear[k], war[k * 96], ep);
      acc = fmaf(w, xp[(size_t)u * 96 + h * 32 + lane] + ep, acc);
    }
    if (d > 0) msum += acc / ssum;
  }
  float val = msum * (1.0f / 3.0f) + cb[lane];
  m_out[(size_t)v * DD + lane] = celuf(val);
}

// fused GRU: one wave -> 16x96 gi and gh via 12 WMMA accumulators, gate math
// in-register, h update + deterministic per-block BN partial sums.
__global__ void __launch_bounds__(256) k_gru(
    const float* __restrict__ Mi, const float* __restrict__ Hold,
    const float* __restrict__ Wi, const float* __restrict__ Wh,
    const float* __restrict__ bi, const float* __restrict__ bh,
    float* __restrict__ Hnew, float* __restrict__ part) {
  int wave = threadIdx.x >> 5, lane = threadIdx.x & 31;
  int half = lane >> 4, l16 = lane & 15;
  int rb = blockIdx.x * 8 + wave;
  bool active = rb < MT;
  v8f aI[6] = {{0},{0},{0},{0},{0},{0}};
  v8f aH[6] = {{0},{0},{0},{0},{0},{0}};
  float s[2] = {0.0f, 0.0f}, sq[2] = {0.0f, 0.0f};
  if (active) {
    const float* mrow = Mi + (size_t)(rb * 16 + l16) * DD + 2 * half;
    const float* hrow = Hold + (size_t)(rb * 16 + l16) * DD + 2 * half;
    for (int k = 0; k < DD; k += 4) {
      v2f am; am.x = mrow[k]; am.y = mrow[k + 1];
      v2f ah; ah.x = hrow[k]; ah.y = hrow[k + 1];
#pragma unroll
      for (int c = 0; c < 6; ++c) {
        const float* bp = Wi + (size_t)(k + 2 * half) * 96 + c * 16 + l16;
        v2f b; b.x = bp[0]; b.y = bp[96];
        aI[c] = wmma4(am, b, aI[c]);
        const float* bq = Wh + (size_t)(k + 2 * half) * 96 + c * 16 + l16;
        v2f b2; b2.x = bq[0]; b2.y = bq[96];
        aH[c] = wmma4(ah, b2, aH[c]);
      }
    }
#pragma unroll
    for (int cc = 0; cc < 2; ++cc) {
      int f = cc * 16 + l16;
      float bir = bi[f], biz = bi[32 + f], bin = bi[64 + f];
      float bhr = bh[f], bhz = bh[32 + f], bhn = bh[64 + f];
#pragma unroll
      for (int j = 0; j < 8; ++j) {
        int n = rb * 16 + j + 8 * half;
        float ir = aI[cc][j] + bir, iz = aI[cc + 2][j] + biz, in_ = aI[cc + 4][j] + bin;
        float hr = aH[cc][j] + bhr, hz = aH[cc + 2][j] + bhz, hn = aH[cc + 4][j] + bhn;
        float r = sigf(ir + hr);
        float z = sigf(iz + hz);
        float nn = tanhf(in_ + r * hn);
        float ho = Hold[(size_t)n * DD + f];
        float hv = (1.0f - z) * nn + z * ho;
        Hnew[(size_t)n * DD + f] = hv;
        s[cc] += hv; sq[cc] += hv * hv;
      }
    }
  }
  __shared__ float lds[8][32][4];
  lds[wave][lane][0] = s[0]; lds[wave][lane][1] = sq[0];
  lds[wave][lane][2] = s[1]; lds[wave][lane][3] = sq[1];
  __syncthreads();
  if (threadIdx.x < 32) {
    int f = threadIdx.x;
    int cc = f >> 4, l = f & 15;
    float S = 0.0f, Q = 0.0f;
    for (int w = 0; w < 8; ++w)
      for (int hh = 0; hh < 2; ++hh) {
        S += lds[w][hh * 16 + l][cc * 2 + 0];
        Q += lds[w][hh * 16 + l][cc * 2 + 1];
      }
    part[blockIdx.x * 64 + f] = S;
    part[blockIdx.x * 64 + 32 + f] = Q;
  }
}

__global__ void k_bnfin(const float* __restrict__ part, float* __restrict__ stat) {
  int f = threadIdx.x;  // 32 threads
  float s = 0.0f, q = 0.0f;
  for (int g = 0; g < NB_GRU; ++g) { s += part[g * 64 + f]; q += part[g * 64 + 32 + f]; }
  float mu = s / (float)NN;
  float var = q / (float)NN - mu * mu;
  stat[f] = mu;
  stat[32 + f] = rsqrtf(var + 1e-5f);
}

__global__ void k_bnapply(const float* __restrict__ h, const float* __restrict__ stat,
                          const float* __restrict__ g, const float* __restrict__ b,
                          float* __restrict__ x) {
  int id = blockIdx.x * blockDim.x + threadIdx.x;
  if (id >= NN * DD) return;
  int f = id & 31;
  x[id] = (h[id] - stat[f]) * stat[32 + f] * g[f] + b[f];
}

__global__ void k_resid(const float* __restrict__ xin, const float* __restrict__ y,
                        const float* __restrict__ lb, float* __restrict__ x) {
  int id = blockIdx.x * blockDim.x + threadIdx.x;
  if (id >= NN * DD) return;
  x[id] = xin[id] + y[id] + lb[id & 31];
}

// ----------------------------- set2set + head -------------------------------
__global__ void k_lstm(float* __restrict__ hl, float* __restrict__ cl,
                       const float* __restrict__ r, const float* __restrict__ wi,
                       const float* __restrict__ wh, const float* __restrict__ b) {
  int bb = blockIdx.x;
  int j = threadIdx.x;  // 0..127
  __shared__ float g[128];
  float acc = b[j];
  for (int k = 0; k < DD; ++k) {
    float hk = hl[bb * DD + k];
    acc = fmaf(hk, wi[k * 128 + j], acc);              // q part of q_star
    acc = fmaf(r[bb * DD + k], wi[(32 + k) * 128 + j], acc);  // r part
    acc = fmaf(hk, wh[k * 128 + j], acc);
  }
  g[j] = acc;
  __syncthreads();
  if (j < DD) {
    float i_ = g[j], f_ = g[32 + j], gg = g[64 + j], o_ = g[96 + j];
    float c = cl[bb * DD + j];
    c = sigf(f_) * c + sigf(i_) * tanhf(gg);
    float h = sigf(o_) * tanhf(c);
    cl[bb * DD + j] = c;
    hl[bb * DD + j] = h;
  }
}

__global__ void __launch_bounds__(256) k_att(const float* __restrict__ x,
                                             const int* __restrict__ gs,
                                             const float* __restrict__ hl,
                                             float* __restrict__ ebuf,
                                             float* __restrict__ r) {
  int b = blockIdx.x, tid = threadIdx.x;
  int st = gs[b], en = gs[b + 1];
  __shared__ float q[32];
  __shared__ float red[256];
  if (tid < 32) q[tid] = hl[b * DD + tid];
  __syncthreads();
  float mx = -3.4e38f;
  for (int n = st + tid; n < en; n += 256) {
    float e = 0.0f;
#pragma unroll
    for (int f = 0; f < DD; ++f) e = fmaf(x[(size_t)n * DD + f], q[f], e);
    ebuf[n] = e;
    mx = fmaxf(mx, e);
  }
  red[tid] = mx;
  __syncthreads();
  for (int ss = 128; ss > 0; ss >>= 1) {
    if (tid < ss) red[tid] = fmaxf(red[tid], red[tid + ss]);
    __syncthreads();
  }
  mx = red[0];
  __syncthreads();
  float sm = 0.0f;
  for (int n = st + tid; n < en; n += 256) {
    float p = expf(ebuf[n] - mx);
    ebuf[n] = p;
    sm += p;
  }
  red[tid] = sm;
  __syncthreads();
  for (int ss = 128; ss > 0; ss >>= 1) {
    if (tid < ss) red[tid] += red[tid + ss];
    __syncthreads();
  }
  float den = red[0];
  __syncthreads();
  int f = tid & 31, w = tid >> 5;
  float acc = 0.0f;
  for (int n = st + w; n < en; n += 8) acc = fmaf(ebuf[n], x[(size_t)n * DD + f], acc);
  red[tid] = acc;
  __syncthreads();
  if (tid < 32) {
    float rs = 0.0f;
    for (int ww = 0; ww < 8; ++ww) rs += red[ww * 32 + f];
    r[b * DD + f] = (den > 0.0f) ? rs / den : 0.0f;
  }
}

__global__ void k_head(const float* __restrict__ hl, const float* __restrict__ r,
                       const float* __restrict__ sw, const float* __restrict__ sb,
                       const float* __restrict__ tw, const float* __restrict__ tb,
                       float* __restrict__ out) {
  int b = blockIdx.x, f = threadIdx.x;  // 32 threads
  __shared__ float g2[32];
  float acc = sb[f];
  for (int k = 0; k < DD; ++k) {
    acc = fmaf(hl[b * DD + k], sw[k * DD + f], acc);
    acc = fmaf(r[b * DD + k], sw[(32 + k) * DD + f], acc);
  }
  g2[f] = celuf(acc);
  __syncthreads();
  if (f < OUTD) {
    float o = tb[f];
    for (int k = 0; k < DD; ++k) o = fmaf(g2[k], tw[k * OUTD + f], o);
    out[b * OUTD + f] = o;
  }
}

// ------------------------------- launcher -----------------------------------
extern "C" void kernel_launch(void* const* d_in, const int* in_sizes, int n_in,
                              void* d_out, int out_size, void* d_ws, size_t ws_size,
                              hipStream_t stream) {
  (void)in_sizes; (void)n_in; (void)out_size; (void)ws_size;
  auto F = [&](int i) { return (const float*)d_in[i]; };
  auto I = [&](int i) { return (const int*)d_in[i]; };
  const float* X0 = F(0);
  const int* EI = I(1);        // [2][E]: row0=src, row1=dst
  const float* EA = F(2);
  const int* BATCH = I(3);
  const float* LIN0_W = F(4);
  const float* LIN0_B = F(5);
  const float* LSTM_WI = F(48);
  const float* LSTM_WH = F(49);
  const float* LSTM_B = F(50);
  const float* SHARE_W = F(51);
  const float* SHARE_B = F(52);
  const float* TASK_W = F(53);
  const float* TASK_B = F(54);
  float* out = (float*)d_out;

  char* pw = (char*)d_ws;
  auto alloc = [&](size_t bytes) {
    void* r = (void*)pw;
    pw += (bytes + 255) & ~(size_t)255;
    return r;
  };
  float* x = (float*)alloc((size_t)NN * DD * 4);
  float* xin = (float*)alloc((size_t)NN * DD * 4);
  float* hA = (float*)alloc((size_t)NN * DD * 4);
  float* hB = (float*)alloc((size_t)NN * DD * 4);
  float* mb = (float*)alloc((size_t)NN * DD * 4);
  float* y = (float*)alloc((size_t)NN * DD * 4);
  float* xp = (float*)alloc((size_t)NN * 96 * 4);
  float* si = (float*)alloc((size_t)NN * HH * 4);
  float* sj = (float*)alloc((size_t)NN * HH * 4);
  float* se = (float*)alloc((size_t)EE * HH * 4);
  float* wae = (float*)alloc(EDDIM * HH * 4);
  float* bnpart = (float*)alloc((size_t)NB_GRU * 64 * 4);
  float* bnstat = (float*)alloc(64 * 4);
  float* ebuf = (float*)alloc((size_t)NN * 4);
  float* s2s = (float*)alloc((size_t)3 * BG * DD * 4);
  float* hl = s2s;
  float* cl = s2s + BG * DD;
  float* rb = s2s + 2 * BG * DD;
  int* deg = (int*)alloc((size_t)NN * 4);
  int* off = (int*)alloc((size_t)(NN + 1) * 4);
  int* cur = (int*)alloc((size_t)NN * 4);
  int* eid = (int*)alloc((size_t)EE * 4);
  int* gs = (int*)alloc(129 * 4);

  auto cdiv = [](int a, int b) { return (a + b - 1) / b; };
  const int GN32 = cdiv(NN * DD, 256);

  // ---- CSR (dst) build + graph offsets (once per call, deterministic) ----
  k_zero_i<<<cdiv(NN, 256), 256, 0, stream>>>(deg, NN);
  k_count<<<cdiv(EE, 256), 256, 0, stream>>>(EI, deg);
  k_scan<<<1, 1024, 0, stream>>>(deg, off, cur);
  k_fill<<<cdiv(EE, 256), 256, 0, stream>>>(EI, cur, eid);
  k_sortcsr<<<cdiv(NN, 256), 256, 0, stream>>>(off, eid);
  k_gstart<<<1, 256, 0, stream>>>(BATCH, gs);

  // ---- lin0: x = celu(X0 @ W + b) ----
  k_gemm<1><<<cdiv(MT * 2, 8), 256, 0, stream>>>(X0, INDIM, LIN0_W, DD, LIN0_B,
                                                 x, DD, MT, 2, INDIM);

  for (int blk = 0; blk < NBLK; ++blk) {
    int pb = 6 + 14 * blk;
    const float* WX = F(pb + 0);
    const float* WE = F(pb + 1);
    const float* AI_ = F(pb + 2);
    const float* AJ_ = F(pb + 3);
    const float* AE_ = F(pb + 4);
    const float* CB = F(pb + 5);
    const float* GWI = F(pb + 6);
    const float* GWH = F(pb + 7);
    const float* GBI = F(pb + 8);
    const float* GBH = F(pb + 9);
    const float* GAM = F(pb + 10);
    const float* BET = F(pb + 11);
    const float* LW = F(pb + 12);
    const float* LB_ = F(pb + 13);

    k_wae<<<1, 64, 0, stream>>>(WE, AE_, wae);
    k_se<<<cdiv(EE * HH, 256), 256, 0, stream>>>(EA, wae, se);
    k_binit<<<GN32, 256, 0, stream>>>(x, xin, hA);
    // y = feats[0] @ lin_w[0:32]
    k_gemm<0><<<cdiv(MT * 2, 8), 256, 0, stream>>>(xin, DD, LW, DD, nullptr,
                                                   y, DD, MT, 2, DD);
    float* hold = hA;
    float* hnew = hB;
    for (int t = 0; t < TT; ++t) {
      k_gemm<0><<<cdiv(MT * 6, 8), 256, 0, stream>>>(x, DD, WX, 96, nullptr,
                                                     xp, 96, MT, 6, DD);
      k_scores<<<cdiv(NN * HH, 256), 256, 0, stream>>>(xp, AI_, AJ_, si, sj);
      k_agg<<<cdiv(NN, 8), 256, 0, stream>>>(EI, off, eid, si, sj, se, xp, EA,
                                             WE, CB, mb);
      k_gru<<<NB_GRU, 256, 0, stream>>>(mb, hold, GWI, GWH, GBI, GBH, hnew, bnpart);
      k_bnfin<<<1, 32, 0, stream>>>(bnpart, bnstat);
      k_bnapply<<<GN32, 256, 0, stream>>>(hnew, bnstat, GAM + t * DD, BET + t * DD, x);
      // y += feats[t+1] @ lin_w[(t+1)*32 : (t+2)*32]
      k_gemm<2><<<cdiv(MT * 2, 8), 256, 0, stream>>>(x, DD, LW + (t + 1) * DD * DD,
                                                     DD, nullptr, y, DD, MT, 2, DD);
      float* tmp = hold; hold = hnew; hnew = tmp;
    }
    k_resid<<<GN32, 256, 0, stream>>>(xin, y, LB_, x);
  }

  // ---- set2set ----
  k_zero_f<<<cdiv(3 * BG * DD, 256), 256, 0, stream>>>(s2s, 3 * BG * DD);
  for (int s = 0; s < 3; ++s) {
    k_lstm<<<BG, 128, 0, stream>>>(hl, cl, rb, LSTM_WI, LSTM_WH, LSTM_B);
    k_att<<<BG, 256, 0, stream>>>(x, gs, hl, ebuf, rb);
  }
  // ---- shared MLP + fused 12 task heads ----
  k_head<<<BG, 32, 0, stream>>>(hl, rb, SHARE_W, SHARE_B, TASK_W, TASK_B, out);
}